// Embedding_87101936763646
// MI455X (gfx1250) — compile-verified
//
#include <hip/hip_runtime.h>
#include <stdint.h>

// Embedding gather via CDNA5 Tensor Data Mover (TDM) gather mode.
// Each block = 1 wave (32 threads). Per block:
//   1) scalar-read 16 token ids (uniform -> SGPRs, branchless fast path)
//   2) tensor_load_to_lds in GATHER mode: 16 rows x 4KB from the embedding
//      table -> 64KB LDS tile (16-bit row indices; VOCAB=50257 < 2^16)
//   3) s_wait_tensorcnt 0
//   4) tensor_store_from_lds (non-temporal): contiguous 64KB tile -> output
// Pure DMA datapath: no VALU on the copy, TENSORcnt tracks completion.
// NT store keeps the ~196MB table resident in the 192MB L2.

typedef uint32_t v4u __attribute__((ext_vector_type(4)));
typedef uint32_t v8u __attribute__((ext_vector_type(8)));

#define ROWS_PER_BLOCK 16
#define D_MODEL 1024
#define VOCAB 50257u

__global__ __launch_bounds__(32) void emb_tdm_gather_kernel(
    const int* __restrict__ X, const float* __restrict__ E,
    float* __restrict__ O, int tokens) {
  __shared__ float tile[ROWS_PER_BLOCK * D_MODEL];  // 64 KB staging tile

  const uint32_t b = blockIdx.x;
  const uint32_t base = b * ROWS_PER_BLOCK;
  int nrow = tokens - (int)base;
  if (nrow > ROWS_PER_BLOCK) nrow = ROWS_PER_BLOCK;

  // Token ids for this tile: uniform address per wave -> scalar loads.
  const int* xp = X + base;
  uint32_t pk[8];
  if (__builtin_expect(nrow == ROWS_PER_BLOCK, 1)) {
    // Full tile: unconditional loads, no branches -> clause-able s_loads.
#pragma unroll
    for (int j = 0; j < 8; ++j) {
      uint32_t lo = (uint32_t)xp[2 * j];
      uint32_t hi = (uint32_t)xp[2 * j + 1];
      pk[j] = (lo & 0xFFFFu) | (hi << 16);
    }
  } else {
    // Tail tile (never taken for tokens % 16 == 0): clamp reads in-bounds.
#pragma unroll
    for (int j = 0; j < 8; ++j) {
      int k0 = 2 * j, k1 = 2 * j + 1;
      uint32_t lo = (uint32_t)xp[k0 < nrow ? k0 : 0];
      uint32_t hi = (uint32_t)xp[k1 < nrow ? k1 : 0];
      if (k0 >= nrow) lo = 0u;
      if (k1 >= nrow) hi = 0u;
      pk[j] = (lo & 0xFFFFu) | (hi << 16);
    }
  }

  // LDS byte offset of the staging tile (low 32 bits of the LDS address).
  const uint32_t lds = (uint32_t)(uintptr_t)(&tile[0]);

  // ---------------- gather-mode load descriptor ----------------
  const uint64_t ea = (uint64_t)(uintptr_t)E;
  // group0: [1:0]=count=1, [30]=0 (16-bit idx), [31]=1 gather_mode,
  //         [63:32]=lds_addr, [120:64]=global_addr, [127:126]=type=2
  v4u g0l = {0x80000001u, lds, (uint32_t)ea,
             ((uint32_t)(ea >> 32) & 0x01FFFFFFu) | 0x80000000u};
  // group1: wg_mask=0, data_size=2 (4B), tensor_dim0=1024, tensor_dim1=VOCAB,
  //         tile_dim0=1024, tile_dim1=#valid indices, dim0_stride=1024
  v8u g1l = {0x00020000u,
             (uint32_t)D_MODEL << 16,     // [63:48] = tensor_dim0 lo16
             (VOCAB & 0xFFFFu) << 16,     // [95:80] = tensor_dim1 lo16
             (uint32_t)D_MODEL << 16,     // [127:112] = tile_dim0
             (uint32_t)nrow,              // [143:128] = tile_dim1 (#indices)
             (uint32_t)D_MODEL,           // [191:160] = dim0_stride lo32
             0u, 0u};
  // groups 2/3: sixteen 16-bit row indices
  v4u g2l = {pk[0], pk[1], pk[2], pk[3]};
  v4u g3l = {pk[4], pk[5], pk[6], pk[7]};

  asm volatile("tensor_load_to_lds %0, %1, %2, %3"
               :
               : "s"(g0l), "s"(g1l), "s"(g2l), "s"(g3l)
               : "memory");

  // Store reads the LDS tile: wait for the gather DMA to land.
  __builtin_amdgcn_s_wait_tensorcnt(0);

  // ---------------- normal-mode store descriptor ----------------
  const uint64_t oa =
      (uint64_t)(uintptr_t)O + (uint64_t)base * (D_MODEL * 4u);
  v4u g0s = {0x00000001u, lds, (uint32_t)oa,
             ((uint32_t)(oa >> 32) & 0x01FFFFFFu) | 0x80000000u};
  v8u g1s = {0x00020000u,
             (uint32_t)D_MODEL << 16,     // tensor_dim0 = 1024
             (uint32_t)nrow << 16,        // tensor_dim1 = nrow
             (uint32_t)D_MODEL << 16,     // tile_dim0 = 1024
             (uint32_t)nrow,              // tile_dim1 = nrow, tile_dim2 unused
             (uint32_t)D_MODEL,           // dim0_stride = 1024
             0u, 0u};
  // group2: tensor_dim2=1, tensor_dim3=1, dim2_stride=0, tile_dim3=0
  v4u g2s = {1u, 1u, 0u, 0u};
  // group3: dim3_stride=0, tensor_dim4=1 ([79:48]), tile_dim4=0
  v4u g3s = {0u, 1u << 16, 0u, 0u};

  // Non-temporal store: output is write-once, keep table rows in L2.
  asm volatile("tensor_store_from_lds %0, %1, %2, %3 th:TH_STORE_NT"
               :
               : "s"(g0s), "s"(g1s), "s"(g2s), "s"(g3s)
               : "memory");

  // Drain before wave exit (S_ENDPGM also waits, but be explicit).
  __builtin_amdgcn_s_wait_tensorcnt(0);
}

extern "C" void kernel_launch(void* const* d_in, const int* in_sizes, int n_in,
                              void* d_out, int out_size, void* d_ws,
                              size_t ws_size, hipStream_t stream) {
  const int* X = (const int*)d_in[0];        // [4, 8192] token ids
  const float* E = (const float*)d_in[1];    // [50257, 1024] table
  float* O = (float*)d_out;                  // [4, 8192, 1024]
  const int tokens = in_sizes[0];            // 32768
  const int blocks = (tokens + ROWS_PER_BLOCK - 1) / ROWS_PER_BLOCK;  // 2048
  emb_tdm_gather_kernel<<<blocks, 32, 0, stream>>>(X, E, O, tokens);
}